// ConditionalDepthwiseConv_38439957299373
// MI455X (gfx1250) — compile-verified
//
#include <hip/hip_runtime.h>
#include <hip/hip_bf16.h>

typedef __attribute__((ext_vector_type(2))) float v2f;
typedef __attribute__((ext_vector_type(8))) float v8f;

// ---------------------------------------------------------------------------
// Problem constants
// ---------------------------------------------------------------------------
#define BB   32
#define CC   64
#define OO   64
#define HH   128
#define WW   128
#define HW   (HH*WW)          // 16384
#define CONDN 16

// workspace layout (float offsets)
#define S_OFF   0             // 32*64*9      = 18432
#define DW_OFF  18432         // 32*576       = 18432
#define PW_OFF  36864         // 32*4096      = 131072
#define DB_OFF  167936        // 32*64        = 2048
#define ST_OFF  169984        // 128 (sum[64], sumsq[64])

// ---------------------------------------------------------------------------
// Kernel 1: S[b,c,kh,kw] = sum_{i,j=0..62} x[b,c,2i+kh,2j+kw]
// (conv3x3 stride2 VALID followed by global mean, reordered)
// One block per (b*64+c) plane.
// ---------------------------------------------------------------------------
__global__ __launch_bounds__(256) void sconv_reduce_kernel(
    const float* __restrict__ x, float* __restrict__ S)
{
    const int bc  = blockIdx.x;
    const int tid = threadIdx.x;
    const float4* xp = reinterpret_cast<const float4*>(x + (size_t)bc * HW);

    float acc[9];
#pragma unroll
    for (int k = 0; k < 9; ++k) acc[k] = 0.f;

    for (int p4 = tid; p4 < HW/4; p4 += 256) {
        const int h  = p4 >> 5;          // 32 float4 per row
        const int wb = (p4 & 31) << 2;
        float4 v = xp[p4];
        float vv[4] = {v.x, v.y, v.z, v.w};
#pragma unroll
        for (int kh = 0; kh < 3; ++kh) {
            const int dh = h - kh;
            const bool okh = (dh >= 0) & (dh <= 124) & ((dh & 1) == 0);
            if (!okh) continue;
#pragma unroll
            for (int e = 0; e < 4; ++e) {
                const int w = wb + e;
#pragma unroll
                for (int kw = 0; kw < 3; ++kw) {
                    const int dw_ = w - kw;
                    const bool ok = (dw_ >= 0) & (dw_ <= 124) & ((dw_ & 1) == 0);
                    if (ok) acc[kh*3 + kw] += vv[e];
                }
            }
        }
    }

    __shared__ float red[256 * 9];
#pragma unroll
    for (int k = 0; k < 9; ++k) red[tid*9 + k] = acc[k];
    __syncthreads();
    for (int s = 128; s > 0; s >>= 1) {
        if (tid < s) {
#pragma unroll
            for (int k = 0; k < 9; ++k) red[tid*9 + k] += red[(tid+s)*9 + k];
        }
        __syncthreads();
    }
    if (tid < 9) S[bc*9 + tid] = red[tid];
}

// ---------------------------------------------------------------------------
// Kernel 2: condition MLP + dynamic weight generation. One block per sample.
// ---------------------------------------------------------------------------
__global__ __launch_bounds__(128) void genparams_kernel(
    const float* __restrict__ S,
    const float* __restrict__ cgw1, const float* __restrict__ cgb1,
    const float* __restrict__ cgw2, const float* __restrict__ cgb2,
    const float* __restrict__ wgw,  const float* __restrict__ wgb,
    const float* __restrict__ pgw,  const float* __restrict__ pgb,
    const float* __restrict__ bgw,  const float* __restrict__ bgb,
    float* __restrict__ dwO, float* __restrict__ pwO, float* __restrict__ dbO)
{
    const int b   = blockIdx.x;
    const int tid = threadIdx.x;
    __shared__ float Ss[576];
    __shared__ float c1[CONDN];
    __shared__ float c2[CONDN];

    for (int i = tid; i < 576; i += 128) Ss[i] = S[b*576 + i];
    __syncthreads();

    if (tid < CONDN) {            // pooled conv + bias, relu
        float a = 0.f;
        const float* w = cgw1 + tid*576;
        for (int i = 0; i < 576; ++i) a += Ss[i] * w[i];
        a = a * (1.0f/3969.0f) + cgb1[tid];
        c1[tid] = fmaxf(a, 0.f);
    }
    __syncthreads();
    if (tid < CONDN) {            // fc 16->16, relu
        float a = cgb2[tid];
        const float* w = cgw2 + tid*CONDN;
#pragma unroll
        for (int k = 0; k < CONDN; ++k) a += c1[k] * w[k];
        c2[tid] = fmaxf(a, 0.f);
    }
    __syncthreads();

    for (int i = tid; i < 576; i += 128) {        // depthwise weights
        float a = wgb[i];
        const float* w = wgw + i*CONDN;
#pragma unroll
        for (int k = 0; k < CONDN; ++k) a += c2[k] * w[k];
        dwO[b*576 + i] = fmaxf(a, 0.f);
    }
    for (int i = tid; i < OO*CC; i += 128) {      // pointwise weights
        float a = pgb[i];
        const float* w = pgw + i*CONDN;
#pragma unroll
        for (int k = 0; k < CONDN; ++k) a += c2[k] * w[k];
        pwO[b*OO*CC + i] = fmaxf(a, 0.f);
    }
    for (int i = tid; i < OO; i += 128) {         // dynamic bias (no relu)
        float a = bgb[i];
        const float* w = bgw + i*CONDN;
#pragma unroll
        for (int k = 0; k < CONDN; ++k) a += c2[k] * w[k];
        dbO[b*OO + i] = a;
    }
}

// ---------------------------------------------------------------------------
// Kernel 3: fused depthwise-conv + per-sample pointwise GEMM via WMMA f32.
// Block = (b, row h, 32-col strip). 256 threads = 8 waves.
// Wave w: M-tile mt = w>>1 (16 o-channels), N-tile nt = w&1 (16 pixels).
// K = channel dim (64), stepped 4 at a time with v_wmma_f32_16x16x4_f32.
// Writes pre-BN result to out.
// ---------------------------------------------------------------------------
#define PW_STRIDE 65          // avoid full-way LDS bank conflicts on A fetch
#define YS_STRIDE 40          // 2*40 mod 64 = 16 -> half-waves hit disjoint banks

__global__ __launch_bounds__(256) void fused_main_kernel(
    const float* __restrict__ x,
    const float* __restrict__ dw, const float* __restrict__ pw,
    const float* __restrict__ db, float* __restrict__ out)
{
    const int b   = blockIdx.z;
    const int h   = blockIdx.y;
    const int w0  = blockIdx.x * 32;
    const int tid = threadIdx.x;

    __shared__ float pwS[OO * PW_STRIDE];   // 4160
    __shared__ float dwS[576];
    __shared__ float dbS[OO];
    __shared__ float xs[64 * 102];          // [c][3 rows][34 cols], 6528
    __shared__ float ys[64 * YS_STRIDE];    // depthwise result [c][32 pix]

    // stage per-sample pointwise matrix (64x64), padded stride
    const float* pwB = pw + b * OO * CC;
    for (int i = tid; i < OO*CC; i += 256) {
        const int o = i >> 6, c = i & 63;
        pwS[o*PW_STRIDE + c] = pwB[i];
    }
    for (int i = tid; i < 576; i += 256) dwS[i] = dw[b*576 + i];
    if (tid < OO) dbS[tid] = db[b*OO + tid];

    // stage x halo: rows h-1..h+1, cols w0-1..w0+32, all 64 channels
    const float* xB = x + (size_t)(b * CC) * HW;
    for (int i = tid; i < 64*102; i += 256) {
        const int c   = i / 102;
        const int rem = i - c*102;
        const int r   = rem / 34;
        const int col = rem - r*34;
        const int hs  = h - 1 + r;
        const int wsrc = w0 - 1 + col;
        float v = 0.f;
        if ((unsigned)hs < (unsigned)HH && (unsigned)wsrc < (unsigned)WW)
            v = xB[c*HW + hs*WW + wsrc];
        xs[i] = v;
    }
    __syncthreads();

    // depthwise 3x3, pad=1: y[c][p], p = col offset within strip
    for (int i = tid; i < 64*32; i += 256) {
        const int c = i >> 5, p = i & 31;
        const float* xc = &xs[c*102 + p];
        const float* wc = &dwS[c*9];
        float a = 0.f;
#pragma unroll
        for (int r = 0; r < 3; ++r)
#pragma unroll
            for (int kw = 0; kw < 3; ++kw)
                a += wc[r*3 + kw] * xc[r*34 + kw];
        ys[c*YS_STRIDE + p] = a;
    }
    __syncthreads();

    // per-sample pointwise GEMM: out[o,n] = sum_c pw[o,c] * y[c,n] + db[o]
    const int wave = tid >> 5;
    const int lane = tid & 31;
    const int mt   = wave >> 1;        // 0..3 -> o block of 16
    const int nt   = wave & 1;         // 0..1 -> pixel block of 16
    const int hi   = lane >> 4;        // half-wave: K pair select
    const int l15  = lane & 15;

    const int o_a = mt*16 + l15;       // A row (o) carried by this lane
    const int n_b = nt*16 + l15;       // B col (pixel) carried by this lane

    v8f acc = {};
#pragma unroll
    for (int kk = 0; kk < 16; ++kk) {
        const int kbase = kk*4 + hi*2;   // lanes 0-15: K=4kk,4kk+1; 16-31: +2,+3
        v2f a, bv;
        a.x  = pwS[o_a*PW_STRIDE + kbase];
        a.y  = pwS[o_a*PW_STRIDE + kbase + 1];
        bv.x = ys[kbase*YS_STRIDE + n_b];
        bv.y = ys[(kbase + 1)*YS_STRIDE + n_b];
        acc = __builtin_amdgcn_wmma_f32_16x16x4_f32(
                  false, a, false, bv, (short)0, acc, false, false);
    }

    // D layout: VGPR r -> M=r (lanes 0-15) / M=r+8 (lanes 16-31); N = lane&15
    float* outB = out + (size_t)(b * OO) * HW + h*WW + w0;
#pragma unroll
    for (int r = 0; r < 8; ++r) {
        const int o = mt*16 + r + hi*8;
        outB[(size_t)o * HW + nt*16 + l15] = acc[r] + dbS[o];
    }
}

// ---------------------------------------------------------------------------
// Kernel 4a: zero the stats accumulators
// ---------------------------------------------------------------------------
__global__ void init_stats_kernel(float* __restrict__ st)
{
    if (threadIdx.x < 128) st[threadIdx.x] = 0.f;
}

// ---------------------------------------------------------------------------
// Kernel 4b: per-channel sum / sumsq over (N,H,W). Block = (o, slice of 4 b).
// ---------------------------------------------------------------------------
__global__ __launch_bounds__(256) void bn_stats_kernel(
    const float* __restrict__ out, float* __restrict__ st)
{
    const int o   = blockIdx.x;   // 0..63
    const int s   = blockIdx.y;   // 0..7
    const int tid = threadIdx.x;
    const float4* o4 = reinterpret_cast<const float4*>(out);

    float sum = 0.f, sq = 0.f;
    for (int i = tid; i < 4*(HW/4); i += 256) {       // 4 samples * 4096 float4
        const int bb  = s*4 + (i >> 12);
        const int hw4 = i & 4095;
        float4 v = o4[((size_t)(bb*OO + o) << 12) + hw4];
        sum += v.x + v.y + v.z + v.w;
        sq  += v.x*v.x + v.y*v.y + v.z*v.z + v.w*v.w;
    }
    __shared__ float s1[256], s2[256];
    s1[tid] = sum; s2[tid] = sq;
    __syncthreads();
    for (int t = 128; t > 0; t >>= 1) {
        if (tid < t) { s1[tid] += s1[tid+t]; s2[tid] += s2[tid+t]; }
        __syncthreads();
    }
    if (tid == 0) {
        atomicAdd(&st[o],      s1[0]);
        atomicAdd(&st[64 + o], s2[0]);
    }
}

// ---------------------------------------------------------------------------
// Kernel 5: apply batchnorm in place (float4).
// ---------------------------------------------------------------------------
__global__ __launch_bounds__(256) void bn_apply_kernel(
    float* __restrict__ out, const float* __restrict__ st,
    const float* __restrict__ gamma, const float* __restrict__ beta)
{
    const float invN = 1.0f / (float)(BB * HW);   // 1/524288
    const size_t idx = (size_t)blockIdx.x * 256 + threadIdx.x;  // float4 index
    const int o = (int)((idx >> 12) & 63);

    const float s1   = st[o];
    const float s2   = st[64 + o];
    const float mean = s1 * invN;
    const float var  = fmaxf(s2 * invN - mean * mean, 0.f);
    const float scl  = gamma[o] * rsqrtf(var + 1e-5f);
    const float sft  = beta[o] - mean * scl;

    float4* p = reinterpret_cast<float4*>(out);
    float4 v = p[idx];
    v.x = v.x * scl + sft; v.y = v.y * scl + sft;
    v.z = v.z * scl + sft; v.w = v.w * scl + sft;
    p[idx] = v;
}

// ---------------------------------------------------------------------------
// Launch
// ---------------------------------------------------------------------------
extern "C" void kernel_launch(void* const* d_in, const int* in_sizes, int n_in,
                              void* d_out, int out_size, void* d_ws, size_t ws_size,
                              hipStream_t stream)
{
    const float* x     = (const float*)d_in[0];
    const float* cgw1  = (const float*)d_in[1];
    const float* cgb1  = (const float*)d_in[2];
    const float* cgw2  = (const float*)d_in[3];
    const float* cgb2  = (const float*)d_in[4];
    const float* wgw   = (const float*)d_in[5];
    const float* wgb   = (const float*)d_in[6];
    const float* pgw   = (const float*)d_in[7];
    const float* pgb   = (const float*)d_in[8];
    const float* bgw   = (const float*)d_in[9];
    const float* bgb   = (const float*)d_in[10];
    const float* gamma = (const float*)d_in[11];
    const float* beta  = (const float*)d_in[12];

    float* ws  = (float*)d_ws;
    float* S   = ws + S_OFF;
    float* dwW = ws + DW_OFF;
    float* pwW = ws + PW_OFF;
    float* dbW = ws + DB_OFF;
    float* st  = ws + ST_OFF;
    float* out = (float*)d_out;

    // 1) strided spatial sums for the condition conv
    sconv_reduce_kernel<<<BB*CC, 256, 0, stream>>>(x, S);

    // 2) condition MLP -> dynamic dw / pw / bias
    genparams_kernel<<<BB, 128, 0, stream>>>(S, cgw1, cgb1, cgw2, cgb2,
                                             wgw, wgb, pgw, pgb, bgw, bgb,
                                             dwW, pwW, dbW);

    // 3) zero BN stat accumulators (independent of 1/2; stream-ordered anyway)
    init_stats_kernel<<<1, 128, 0, stream>>>(st);

    // 4) fused depthwise + WMMA pointwise GEMM, pre-BN result into d_out
    dim3 grid(WW/32, HH, BB);
    fused_main_kernel<<<grid, 256, 0, stream>>>(x, dwW, pwW, dbW, out);

    // 5) per-channel batch statistics
    bn_stats_kernel<<<dim3(OO, 8), 256, 0, stream>>>(out, st);

    // 6) normalize in place
    bn_apply_kernel<<<(BB*OO*HW/4 + 255)/256, 256, 0, stream>>>(out, st, gamma, beta);
}